// DistributedInverseRealVectorSHT_17549236371794
// MI455X (gfx1250) — compile-verified
//
#include <hip/hip_runtime.h>

// DistributedInverseRealVectorSHT for MI455X (gfx1250), all-f32 WMMA path.
//
// Stage 1 (Legendre): per-m GEMMs via V_WMMA_F32_16X16X4_F32, 16 accumulator
//   chains (xr0/xi0/xr1/xi1 x d0/d1 x 2 k-columns), VALU-combined into
//   srl/sim/trl/tim (f32 WMMA forbids A/B negation per ISA NEG rules).
// Stage 2 (irfft, norm='forward' == unscaled real inverse DFT): GEMM against a
//   precomputed [cos | -sin] weight table with exact integer phase reduction.
//
// All bounds handling is branch-free: main loops are guard-free, tails use
// clamped addresses + a 0/1 lane mask multiply.

typedef float v2f __attribute__((ext_vector_type(2)));
typedef float v8f __attribute__((ext_vector_type(8)));

#define SHT_LMAX 361
#define SHT_MMAX 361
#define SHT_NLAT 361
#define SHT_NLON 720
#define SHT_NCH  32
#define SHT_KTOT 722                 // 2*MMAX: Re rows then Im rows
#define SHT_LL2  (361 * 361)

#define WMMA_F32(A, B, C) \
  __builtin_amdgcn_wmma_f32_16x16x4_f32(false, (A), false, (B), (short)0, (C), false, false)

// ---------------------------------------------------------------------------
// Twiddle table: wtab[K][n], K in [0,722), n in [0,720)
//   K <  361 : w_m *  cos(2*pi*m*n/720)   (m = K)
//   K >= 361 : w_m * -sin(2*pi*m*n/720)   (m = K-361)
// w_m = 1 for m==0 or m==360 (DC / Nyquist counted once), else 2.
// ---------------------------------------------------------------------------
__global__ void sht_twiddle_kernel(float* __restrict__ wtab) {
  const int idx = blockIdx.x * blockDim.x + threadIdx.x;
  if (idx >= SHT_KTOT * SHT_NLON) return;
  const int K = idx / SHT_NLON;
  const int n = idx % SHT_NLON;
  const int m = (K < SHT_MMAX) ? K : (K - SHT_MMAX);
  const float w = (m == 0 || m == SHT_MMAX - 1) ? 1.0f : 2.0f;
  const int phase = (int)(((long long)m * (long long)n) % SHT_NLON);
  const float theta = (float)phase * (6.283185307179586476925287e0f / (float)SHT_NLON);
  wtab[idx] = (K < SHT_MMAX) ? (w * cosf(theta)) : (-w * sinf(theta));
}

// ---------------------------------------------------------------------------
// Stage 1: for fixed m, D-tile[c(16) x k(32)] = sum_l A[c][l] * d[m][k][l]
// grid = (k-supertiles=12, m=361, c-tiles=2), block = 1 wave (32 lanes).
// A fragment (16x4 f32): lane L holds M=L%16, K = {0,1} (L<16) / {2,3} (L>=16)
// B fragment (4x16 f32): lane L holds N=L%16, same K grouping
// D tile (16x16 f32):    VGPR v, lane L -> row v + 8*(L/16), col L%16
// ---------------------------------------------------------------------------
__global__ __launch_bounds__(32)
void sht_legendre_wmma(const float* __restrict__ x_re,
                       const float* __restrict__ x_im,
                       const float* __restrict__ d0,
                       const float* __restrict__ d1,
                       float* __restrict__ sbuf) {
  const int kst   = blockIdx.x;           // 0..11 (32-wide k supertile)
  const int m     = blockIdx.y;           // 0..360
  const int ctile = blockIdx.z;           // 0..1
  const int lane  = threadIdx.x;
  const int half  = lane >> 4;
  const int sub   = lane & 15;
  const int k0 = kst * 32;
  const int c0 = ctile * 16;
  const int lh = 2 * half;

  const int c  = c0 + sub;                // A row channel (< 32)
  const int kA = k0 + sub;                // B columns (nlat); may exceed 360,
  const int kB = kA + 16;                 // clamped loads feed discarded cols.
  const int kAs = (kA < SHT_NLAT) ? kA : (SHT_NLAT - 1);
  const int kBs = (kB < SHT_NLAT) ? kB : (SHT_NLAT - 1);

  // x layout: x[((c*2 + ri)*LMAX + l)*MMAX + m]
  const size_t xb0 = ((size_t)(c * 2 + 0) * SHT_LMAX) * SHT_MMAX + (size_t)m;
  const size_t xb1 = ((size_t)(c * 2 + 1) * SHT_LMAX) * SHT_MMAX + (size_t)m;
  // d layout: d[(m*NLAT + k)*LMAX + l]
  const size_t dbA = ((size_t)m * SHT_NLAT + (size_t)kAs) * SHT_LMAX;
  const size_t dbB = ((size_t)m * SHT_NLAT + (size_t)kBs) * SHT_LMAX;

  v8f A_r0d0 = {}, A_r0d1 = {}, A_r1d0 = {}, A_r1d1 = {};
  v8f A_i0d0 = {}, A_i0d1 = {}, A_i1d0 = {}, A_i1d1 = {};
  v8f B_r0d0 = {}, B_r0d1 = {}, B_r1d0 = {}, B_r1d1 = {};
  v8f B_i0d0 = {}, B_i0d1 = {}, B_i1d0 = {}, B_i1d1 = {};

  // Main loop: l in [0,360) -- both lane halves fully in range, no guards.
  for (int l0 = 0; l0 < 360; l0 += 4) {
    const size_t la = (size_t)(l0 + lh);
    v2f fr0, fr1, fi0, fi1, dA0, dA1, dB0, dB1;
    fr0.x = x_re[xb0 + la * SHT_MMAX];
    fr0.y = x_re[xb0 + (la + 1) * SHT_MMAX];
    fi0.x = x_im[xb0 + la * SHT_MMAX];
    fi0.y = x_im[xb0 + (la + 1) * SHT_MMAX];
    fr1.x = x_re[xb1 + la * SHT_MMAX];
    fr1.y = x_re[xb1 + (la + 1) * SHT_MMAX];
    fi1.x = x_im[xb1 + la * SHT_MMAX];
    fi1.y = x_im[xb1 + (la + 1) * SHT_MMAX];
    dA0.x = __builtin_nontemporal_load(&d0[dbA + la]);
    dA0.y = __builtin_nontemporal_load(&d0[dbA + la + 1]);
    dA1.x = __builtin_nontemporal_load(&d1[dbA + la]);
    dA1.y = __builtin_nontemporal_load(&d1[dbA + la + 1]);
    dB0.x = __builtin_nontemporal_load(&d0[dbB + la]);
    dB0.y = __builtin_nontemporal_load(&d0[dbB + la + 1]);
    dB1.x = __builtin_nontemporal_load(&d1[dbB + la]);
    dB1.y = __builtin_nontemporal_load(&d1[dbB + la + 1]);

    A_r0d0 = WMMA_F32(fr0, dA0, A_r0d0);
    A_i1d1 = WMMA_F32(fi1, dA1, A_i1d1);
    A_i0d0 = WMMA_F32(fi0, dA0, A_i0d0);
    A_r1d1 = WMMA_F32(fr1, dA1, A_r1d1);
    A_i0d1 = WMMA_F32(fi0, dA1, A_i0d1);
    A_r1d0 = WMMA_F32(fr1, dA0, A_r1d0);
    A_r0d1 = WMMA_F32(fr0, dA1, A_r0d1);
    A_i1d0 = WMMA_F32(fi1, dA0, A_i1d0);
    B_r0d0 = WMMA_F32(fr0, dB0, B_r0d0);
    B_i1d1 = WMMA_F32(fi1, dB1, B_i1d1);
    B_i0d0 = WMMA_F32(fi0, dB0, B_i0d0);
    B_r1d1 = WMMA_F32(fr1, dB1, B_r1d1);
    B_i0d1 = WMMA_F32(fi0, dB1, B_i0d1);
    B_r1d0 = WMMA_F32(fr1, dB0, B_r1d0);
    B_r0d1 = WMMA_F32(fr0, dB1, B_r0d1);
    B_i1d0 = WMMA_F32(fi1, dB0, B_i1d0);
  }

  // Tail: l = 360. Only lane-half 0, K-slot 0 is live; loads use the valid
  // l=360 address for every lane and the dead half is zeroed by mask multiply.
  {
    const float mH = (half == 0) ? 1.0f : 0.0f;
    const size_t lt = (size_t)360;
    v2f fr0 = { x_re[xb0 + lt * SHT_MMAX], 0.0f };
    v2f fi0 = { x_im[xb0 + lt * SHT_MMAX], 0.0f };
    v2f fr1 = { x_re[xb1 + lt * SHT_MMAX], 0.0f };
    v2f fi1 = { x_im[xb1 + lt * SHT_MMAX], 0.0f };
    v2f dA0 = { __builtin_nontemporal_load(&d0[dbA + 360]) * mH, 0.0f };
    v2f dA1 = { __builtin_nontemporal_load(&d1[dbA + 360]) * mH, 0.0f };
    v2f dB0 = { __builtin_nontemporal_load(&d0[dbB + 360]) * mH, 0.0f };
    v2f dB1 = { __builtin_nontemporal_load(&d1[dbB + 360]) * mH, 0.0f };

    A_r0d0 = WMMA_F32(fr0, dA0, A_r0d0);
    A_i1d1 = WMMA_F32(fi1, dA1, A_i1d1);
    A_i0d0 = WMMA_F32(fi0, dA0, A_i0d0);
    A_r1d1 = WMMA_F32(fr1, dA1, A_r1d1);
    A_i0d1 = WMMA_F32(fi0, dA1, A_i0d1);
    A_r1d0 = WMMA_F32(fr1, dA0, A_r1d0);
    A_r0d1 = WMMA_F32(fr0, dA1, A_r0d1);
    A_i1d0 = WMMA_F32(fi1, dA0, A_i1d0);
    B_r0d0 = WMMA_F32(fr0, dB0, B_r0d0);
    B_i1d1 = WMMA_F32(fi1, dB1, B_i1d1);
    B_i0d0 = WMMA_F32(fi0, dB0, B_i0d0);
    B_r1d1 = WMMA_F32(fr1, dB1, B_r1d1);
    B_i0d1 = WMMA_F32(fi0, dB1, B_i0d1);
    B_r1d0 = WMMA_F32(fr1, dB0, B_r1d0);
    B_r0d1 = WMMA_F32(fr0, dB1, B_r0d1);
    B_i1d0 = WMMA_F32(fi1, dB0, B_i1d0);
  }

  // Combine (f32 WMMA cannot negate A/B operands -> VALU combine):
  const v8f srlA = A_r0d0 - A_i1d1;
  const v8f simA = A_i0d0 + A_r1d1;
  const v8f trlA = -(A_i0d1 + A_r1d0);
  const v8f timA = A_r0d1 - A_i1d0;
  const v8f srlB = B_r0d0 - B_i1d1;
  const v8f simB = B_i0d0 + B_r1d1;
  const v8f trlB = -(B_i0d1 + B_r1d0);
  const v8f timB = B_r0d1 - B_i1d0;

  // sbuf layout: S[((c*2+st)*2 + rc) * LL2 + m*NLAT + k]  (k contiguous ->
  // coalesced stores matching the D-tile lane mapping).
#pragma unroll
  for (int v = 0; v < 8; ++v) {
    const int cc = c0 + v + 8 * half;
    const size_t row = (size_t)(cc * 4) * SHT_LL2 + (size_t)m * SHT_NLAT;
    if (kA < SHT_NLAT) {
      sbuf[row + (size_t)kA]                       = srlA[v];
      sbuf[row + (size_t)SHT_LL2 + (size_t)kA]     = simA[v];
      sbuf[row + (size_t)2 * SHT_LL2 + (size_t)kA] = trlA[v];
      sbuf[row + (size_t)3 * SHT_LL2 + (size_t)kA] = timA[v];
    }
    if (kB < SHT_NLAT) {
      sbuf[row + (size_t)kB]                       = srlB[v];
      sbuf[row + (size_t)SHT_LL2 + (size_t)kB]     = simB[v];
      sbuf[row + (size_t)2 * SHT_LL2 + (size_t)kB] = trlB[v];
      sbuf[row + (size_t)3 * SHT_LL2 + (size_t)kB] = timB[v];
    }
  }
}

// ---------------------------------------------------------------------------
// Stage 2: out[cs][k][n] = sum_{K=0..721} S[cs][K][k] * wtab[K][n]
// (K = rc*361 + m concatenates the Re and Im contractions in one chain.)
// grid = (n-supertiles=15 (48-wide), k-supertiles=12 (32-wide), cs=64).
// ---------------------------------------------------------------------------
__global__ __launch_bounds__(32)
void sht_idft_wmma(const float* __restrict__ sbuf,
                   const float* __restrict__ wtab,
                   float* __restrict__ out) {
  const int nst = blockIdx.x;             // 0..14
  const int kst = blockIdx.y;             // 0..11
  const int cs  = blockIdx.z;             // 0..63  (c*2 + st)
  const int lane = threadIdx.x;
  const int half = lane >> 4;
  const int sub  = lane & 15;
  const int n0 = nst * 48;
  const int k0 = kst * 32;
  const int lh = 2 * half;

  const int kA = k0 + sub;                // A rows (nlat); clamped loads feed
  const int kB = kA + 16;                 // discarded D rows when >= 361.
  const int kAs = (kA < SHT_NLAT) ? kA : (SHT_NLAT - 1);
  const int kBs = (kB < SHT_NLAT) ? kB : (SHT_NLAT - 1);
  const int nc0 = n0 + sub;               // B columns, always < 720
  const int nc1 = nc0 + 16;
  const int nc2 = nc0 + 32;

  const float* __restrict__ sb = sbuf + (size_t)cs * 2 * SHT_LL2;

  v8f a00 = {}, a01 = {}, a02 = {}, a10 = {}, a11 = {}, a12 = {};

  // Main loop: K in [0,720) -- both lane halves fully in range.
  for (int K0 = 0; K0 < 720; K0 += 4) {
    const size_t Ka = (size_t)(K0 + lh);
    v2f fA, fB, w0, w1, w2;
    fA.x = sb[Ka * SHT_NLAT + (size_t)kAs];
    fA.y = sb[(Ka + 1) * SHT_NLAT + (size_t)kAs];
    fB.x = sb[Ka * SHT_NLAT + (size_t)kBs];
    fB.y = sb[(Ka + 1) * SHT_NLAT + (size_t)kBs];
    w0.x = wtab[Ka * SHT_NLON + (size_t)nc0];
    w0.y = wtab[(Ka + 1) * SHT_NLON + (size_t)nc0];
    w1.x = wtab[Ka * SHT_NLON + (size_t)nc1];
    w1.y = wtab[(Ka + 1) * SHT_NLON + (size_t)nc1];
    w2.x = wtab[Ka * SHT_NLON + (size_t)nc2];
    w2.y = wtab[(Ka + 1) * SHT_NLON + (size_t)nc2];

    a00 = WMMA_F32(fA, w0, a00);
    a01 = WMMA_F32(fA, w1, a01);
    a02 = WMMA_F32(fA, w2, a02);
    a10 = WMMA_F32(fB, w0, a10);
    a11 = WMMA_F32(fB, w1, a11);
    a12 = WMMA_F32(fB, w2, a12);
  }

  // Tail: K in {720,721}, live only for lane-half 0; mask A side to zero.
  {
    const float mH = (half == 0) ? 1.0f : 0.0f;
    v2f fA, fB, w0, w1, w2;
    fA.x = sb[(size_t)720 * SHT_NLAT + (size_t)kAs] * mH;
    fA.y = sb[(size_t)721 * SHT_NLAT + (size_t)kAs] * mH;
    fB.x = sb[(size_t)720 * SHT_NLAT + (size_t)kBs] * mH;
    fB.y = sb[(size_t)721 * SHT_NLAT + (size_t)kBs] * mH;
    w0.x = wtab[(size_t)720 * SHT_NLON + (size_t)nc0];
    w0.y = wtab[(size_t)721 * SHT_NLON + (size_t)nc0];
    w1.x = wtab[(size_t)720 * SHT_NLON + (size_t)nc1];
    w1.y = wtab[(size_t)721 * SHT_NLON + (size_t)nc1];
    w2.x = wtab[(size_t)720 * SHT_NLON + (size_t)nc2];
    w2.y = wtab[(size_t)721 * SHT_NLON + (size_t)nc2];

    a00 = WMMA_F32(fA, w0, a00);
    a01 = WMMA_F32(fA, w1, a01);
    a02 = WMMA_F32(fA, w2, a02);
    a10 = WMMA_F32(fB, w0, a10);
    a11 = WMMA_F32(fB, w1, a11);
    a12 = WMMA_F32(fB, w2, a12);
  }

  // Stores: n contiguous across lanes, nontemporal (output is never re-read).
#pragma unroll
  for (int v = 0; v < 8; ++v) {
    const int rA = k0 + v + 8 * half;
    const int rB = rA + 16;
    if (rA < SHT_NLAT) {
      const size_t o = ((size_t)cs * SHT_NLAT + (size_t)rA) * SHT_NLON;
      __builtin_nontemporal_store(a00[v], &out[o + (size_t)nc0]);
      __builtin_nontemporal_store(a01[v], &out[o + (size_t)nc1]);
      __builtin_nontemporal_store(a02[v], &out[o + (size_t)nc2]);
    }
    if (rB < SHT_NLAT) {
      const size_t o = ((size_t)cs * SHT_NLAT + (size_t)rB) * SHT_NLON;
      __builtin_nontemporal_store(a10[v], &out[o + (size_t)nc0]);
      __builtin_nontemporal_store(a11[v], &out[o + (size_t)nc1]);
      __builtin_nontemporal_store(a12[v], &out[o + (size_t)nc2]);
    }
  }
}

// ---------------------------------------------------------------------------
extern "C" void kernel_launch(void* const* d_in, const int* in_sizes, int n_in,
                              void* d_out, int out_size, void* d_ws, size_t ws_size,
                              hipStream_t stream) {
  (void)in_sizes; (void)n_in; (void)out_size; (void)ws_size;

  const float* x_re = (const float*)d_in[0];  // (1,32,2,361,361)
  const float* x_im = (const float*)d_in[1];  // (1,32,2,361,361)
  const float* d0   = (const float*)d_in[2];  // (361,361,361)
  const float* d1   = (const float*)d_in[3];  // (361,361,361)
  float* out = (float*)d_out;                 // (1,32,2,361,720)

  float* ws   = (float*)d_ws;
  float* sbuf = ws;                                        // 32*4*361*361 floats
  float* wtab = ws + (size_t)SHT_NCH * 4 * SHT_LL2;        // 722*720 floats

  // Twiddle table (722*720 = 519,840 elems)
  const int tw = SHT_KTOT * SHT_NLON;
  sht_twiddle_kernel<<<(tw + 255) / 256, 256, 0, stream>>>(wtab);

  // Stage 1: Legendre contractions
  dim3 g1(12, SHT_MMAX, 2);
  sht_legendre_wmma<<<g1, 32, 0, stream>>>(x_re, x_im, d0, d1, sbuf);

  // Stage 2: inverse real DFT
  dim3 g2(15, 12, 64);
  sht_idft_wmma<<<g2, 32, 0, stream>>>(sbuf, wtab, out);
}